// StDevLoss_69372311765092
// MI455X (gfx1250) — compile-verified
//
#include <hip/hip_runtime.h>
#include <stdint.h>

// Problem geometry (fixed by the reference: depth is (1,1,480,640) f32)
#define ROWS 480
#define COLS 640
#define TS   16            // output tile = 16x16 pixels per 256-thread block
#define TMAX 28            // staged region max extent: TS + 2*6 halo
#define NBX  (COLS / TS)   // 40
#define NBY  (ROWS / TS)   // 30

#define FXc 582.6244816773795f
#define FYc 582.6910327098864f
#define CXc 313.0447587080473f
#define CYc 238.44389626620386f

#if defined(__AMDGCN__) && __has_builtin(__builtin_amdgcn_tensor_load_to_lds) && \
    __has_builtin(__builtin_amdgcn_s_wait_tensorcnt)
#define USE_TDM 1
#else
#define USE_TDM 0
#endif

typedef float        v2f   __attribute__((ext_vector_type(2)));
typedef float        v8f   __attribute__((ext_vector_type(8)));
typedef unsigned int u32x4 __attribute__((ext_vector_type(4)));
typedef int          i32x4 __attribute__((ext_vector_type(4)));
typedef int          i32x8 __attribute__((ext_vector_type(8)));

__device__ __forceinline__ int imin(int a, int b) { return a < b ? a : b; }
__device__ __forceinline__ int imax(int a, int b) { return a > b ? a : b; }

// Raw v_sqrt_f32 (~1 ulp): all inputs here are far from the denormal range,
// so we can skip the IEEE scaling fix-up ladder the libm sqrtf emits.
__device__ __forceinline__ float fast_sqrtf(float x) {
#if defined(__AMDGCN__) && __has_builtin(__builtin_amdgcn_sqrtf)
  return __builtin_amdgcn_sqrtf(x);
#else
  return sqrtf(x);
#endif
}

// ---------------------------------------------------------------------------
// Full-wave32 sum of one float per lane.
// WMMA path: D = A x Ones + 0 with V_WMMA_F32_16X16X4_F32.
//   A layout (ISA 7.12.2, 32-bit A 16x4): lane m (0-15): VGPR0=K0, VGPR1=K1;
//   lane m+16: VGPR0=K2, VGPR1=K3.  With a=(acc,0): rowsum(m)=acc_m+acc_{m+16}.
//   D layout: lanes 0-15 VGPR v = D[v,lane] = rowsum(v); lanes 16-31 = rowsum(v+8).
//   Per-lane sum of 8 VGPRs then a SWAPX16 swizzle-add gives the full wave sum.
// B = all-ones is layout-invariant, so this is correct by the documented tables.
// ---------------------------------------------------------------------------
__device__ __forceinline__ float wave_sum(float v) {
#if defined(__AMDGCN__) && __has_builtin(__builtin_amdgcn_wmma_f32_16x16x4_f32)
  v2f a; a[0] = v;    a[1] = 0.0f;
  v2f b; b[0] = 1.0f; b[1] = 1.0f;
  v8f c = {};
  v8f d = __builtin_amdgcn_wmma_f32_16x16x4_f32(
      /*neg_a=*/false, a, /*neg_b=*/false, b,
      /*c_mod=*/(short)0, c, /*reuse_a=*/false, /*reuse_b=*/false);
  float s = ((d[0] + d[1]) + (d[2] + d[3])) + ((d[4] + d[5]) + (d[6] + d[7]));
  // add the other half-wave's partial (SWAPX16: xor=0x10,and=0x1f)
  s += __int_as_float(__builtin_amdgcn_ds_swizzle(__float_as_int(s), 0x401F));
  return s;
#else
  float s = v;  // butterfly via ds_swizzle xor patterns (group-of-32)
  s += __int_as_float(__builtin_amdgcn_ds_swizzle(__float_as_int(s), 0x041F)); // xor 1
  s += __int_as_float(__builtin_amdgcn_ds_swizzle(__float_as_int(s), 0x081F)); // xor 2
  s += __int_as_float(__builtin_amdgcn_ds_swizzle(__float_as_int(s), 0x101F)); // xor 4
  s += __int_as_float(__builtin_amdgcn_ds_swizzle(__float_as_int(s), 0x201F)); // xor 8
  s += __int_as_float(__builtin_amdgcn_ds_swizzle(__float_as_int(s), 0x401F)); // xor 16
  return s;
#endif
}

// ---------------------------------------------------------------------------
// Kernel 1: one 16x16 output tile per block (256 threads = 8 wave32).
// Stage the clamped 28x28 halo region of depth into LDS via the Tensor Data
// Mover (2D D#), convert depth -> masked z in place, then each thread does its
// 7x7 clamped-window std.  Block partial sums -> d_ws.
// ---------------------------------------------------------------------------
__global__ __launch_bounds__(256) void stdev_tile_kernel(
    const float* __restrict__ depth, float* __restrict__ bsum) {
  __shared__ float tile[TMAX * TMAX];
  __shared__ float wpart[8];

  const int bx = blockIdx.x, by = blockIdx.y;
  const int r0 = by * TS, c0 = bx * TS;
  // Clamped windows always stay inside [i-6, i+6] of the image, so a 6-halo
  // clipped to the image covers every window of this tile.
  const int rlo = imax(r0 - 6, 0), rhi = imin(r0 + TS + 6, ROWS);
  const int clo = imax(c0 - 6, 0), chi = imin(c0 + TS + 6, COLS);
  const int H = rhi - rlo, W = chi - clo;   // 22..28
  const int tid = threadIdx.x;

#if USE_TDM
  if (tid < 32) {  // wave 0 issues the DMA (EXEC ignored by TDM; wave-uniform)
    const uint32_t lds_off = (uint32_t)(uintptr_t)&tile[0];
    const uint64_t gaddr =
        (uint64_t)(uintptr_t)depth + 4ull * ((uint64_t)rlo * COLS + (uint64_t)clo);
    // D# group 0: count=1 | lds_addr | global_addr[56:0] | type=2
    u32x4 g0;
    g0[0] = 1u;
    g0[1] = lds_off;
    g0[2] = (uint32_t)(gaddr & 0xffffffffu);
    g0[3] = ((uint32_t)((gaddr >> 32) & 0x01ffffffu)) | (2u << 30);
    // D# group 1: data_size=4B; tensor 640x480 stride 640; tile W x H
    i32x8 g1;
    g1[0] = (int)(2u << 16);                                  // data_size=2 (4B)
    g1[1] = (int)(((uint32_t)COLS) << 16);                    // tensor_dim0 lo16
    g1[2] = (int)(((uint32_t)ROWS) << 16);                    // dim0 hi=0 | tensor_dim1 lo16
    g1[3] = (int)(((uint32_t)W) << 16);                       // dim1 hi=0 | tile_dim0=W
    g1[4] = (int)(uint32_t)H;                                 // tile_dim1=H | tile_dim2=0
    g1[5] = COLS;                                             // tensor_dim0_stride lo32
    g1[6] = (int)((((uint32_t)(ROWS * COLS)) & 0xffffu) << 16); // stride0 hi | stride1 lo16
    g1[7] = (int)(((uint32_t)(ROWS * COLS)) >> 16);           // stride1 [47:16]
    i32x4 g2; g2[0] = 1; g2[1] = 0; g2[2] = 0; g2[3] = 0;     // tensor_dim2=1, rest 0
    i32x4 g3; g3[0] = 0; g3[1] = 0; g3[2] = 0; g3[3] = 0;
    // 6-arg toolchain form: 5th operand models the unused VADDR4 SGPR group
    // (ISA 7.3) -> zero-filled.
    i32x8 g4;
    g4[0] = 0; g4[1] = 0; g4[2] = 0; g4[3] = 0;
    g4[4] = 0; g4[5] = 0; g4[6] = 0; g4[7] = 0;
    __builtin_amdgcn_tensor_load_to_lds(g0, g1, g2, g3, g4, 0);
    __builtin_amdgcn_s_wait_tensorcnt(0);
  }
  __syncthreads();
#else
  for (int i = tid; i < H * W; i += 256) {
    const int r = i / W;
    const int c = i - r * W;
    tile[i] = depth[(size_t)(rlo + r) * COLS + (clo + c)];
  }
  __syncthreads();
#endif

  // depth -> masked z in place (one divide per staged element, not per window elem)
  for (int i = tid; i < H * W; i += 256) {
    const float dv = tile[i];
    tile[i] = (dv > 0.0f && dv < 1.01f) ? (dv / 1000.0f) : 0.0f;
  }
  __syncthreads();

  const int tx = tid & (TS - 1), ty = tid >> 4;
  const int gc = c0 + tx, gr = r0 + ty;
  const int istart = imin(imax(gr - 3, 0), ROWS - 7);
  const int jstart = imin(imax(gc - 3, 0), COLS - 7);

  const float zc = tile[(gr - rlo) * W + (gc - clo)];
  const float xc = zc * (((float)gc - CXc) / FXc);
  const float yc = zc * (((float)gr - CYc) / FYc);

  float cfac[7], rfac[7];
#pragma unroll
  for (int k = 0; k < 7; ++k) {
    cfac[k] = ((float)(jstart + k) - CXc) / FXc;
    rfac[k] = ((float)(istart + k) - CYc) / FYc;
  }

  float sum = 0.0f, sumsq = 0.0f;
#pragma unroll
  for (int di = 0; di < 7; ++di) {
    const float* row = &tile[(istart + di - rlo) * W + (jstart - clo)];
    const float yf = rfac[di];
#pragma unroll
    for (int dj = 0; dj < 7; ++dj) {
      const float zw  = row[dj];
      const float xw  = zw * cfac[dj];
      const float yw  = zw * yf;
      const float ddx = xc - xw, ddy = yc - yw, ddz = zc - zw;
      const float dist = fast_sqrtf(ddx * ddx + ddy * ddy + ddz * ddz);
      sum   += dist;
      sumsq += dist * dist;
    }
  }
  const float mean = sum * (1.0f / 49.0f);
  const float var  = fmaxf(sumsq - sum * mean, 0.0f) * (1.0f / 48.0f);
  const float dev  = (zc > 0.0f) ? fast_sqrtf(var) : 0.0f;  // zc>0 <=> center valid

  // block reduction: WMMA wave sum, then 8 wave partials via LDS (deterministic)
  const float wsv = wave_sum(dev);
  if ((tid & 31) == 0) wpart[tid >> 5] = wsv;
  __syncthreads();
  if (tid == 0) {
    const float t = ((wpart[0] + wpart[1]) + (wpart[2] + wpart[3])) +
                    ((wpart[4] + wpart[5]) + (wpart[6] + wpart[7]));
    bsum[by * NBX + bx] = t;
  }
}

// ---------------------------------------------------------------------------
// Kernel 2: single-block deterministic final reduction of the 1200 block sums.
// ---------------------------------------------------------------------------
__global__ __launch_bounds__(256) void final_reduce_kernel(
    const float* __restrict__ bsum, int n, float* __restrict__ out) {
  __shared__ float wpart[8];
  const int tid = threadIdx.x;
  float s = 0.0f;
  for (int i = tid; i < n; i += 256) s += bsum[i];  // fixed order -> deterministic
  const float wsv = wave_sum(s);
  if ((tid & 31) == 0) wpart[tid >> 5] = wsv;
  __syncthreads();
  if (tid == 0) {
    const float t = ((wpart[0] + wpart[1]) + (wpart[2] + wpart[3])) +
                    ((wpart[4] + wpart[5]) + (wpart[6] + wpart[7]));
    out[0] = t * 100.0f;
  }
}

extern "C" void kernel_launch(void* const* d_in, const int* in_sizes, int n_in,
                              void* d_out, int out_size, void* d_ws, size_t ws_size,
                              hipStream_t stream) {
  const float* depth = (const float*)d_in[0];   // (1,1,480,640) f32
  // d_in[1] = gt, unused by the reference's hot path
  float* out  = (float*)d_out;
  float* bsum = (float*)d_ws;                   // NBX*NBY = 1200 floats of scratch

  dim3 grid(NBX, NBY);
  stdev_tile_kernel<<<grid, 256, 0, stream>>>(depth, bsum);
  final_reduce_kernel<<<1, 256, 0, stream>>>(bsum, NBX * NBY, out);
}